// YOLOv4Loss_66563403153769
// MI455X (gfx1250) — compile-verified
//
#include <hip/hip_runtime.h>
#include <math.h>

#define NB 16
#define NA 3
#define NT 50
#define NC 80
#define CH 85      // 5 + NC
#define IMGF 608.0f

typedef float v2f __attribute__((ext_vector_type(2)));
typedef float v8f __attribute__((ext_vector_type(8)));

__constant__ float c_anchors[3][3][2] = {
    {{12.f,16.f},{19.f,36.f},{40.f,28.f}},
    {{36.f,75.f},{76.f,55.f},{72.f,146.f}},
    {{142.f,110.f},{192.f,243.f},{459.f,401.f}}
};

__device__ __forceinline__ float clamp01(float x) { return fminf(fmaxf(x, 0.f), 1.f); }
__device__ __forceinline__ float sigf(float x)    { return 1.f / (1.f + expf(-x)); }

__device__ __forceinline__ float iou_ccwh(float ax, float ay, float aw, float ah,
                                          float bx, float by, float bw, float bh) {
    float ax1 = ax - aw * 0.5f, ay1 = ay - ah * 0.5f;
    float ax2 = ax + aw * 0.5f, ay2 = ay + ah * 0.5f;
    float bx1 = bx - bw * 0.5f, by1 = by - bh * 0.5f;
    float bx2 = bx + bw * 0.5f, by2 = by + bh * 0.5f;
    float iw = fmaxf(fminf(ax2, bx2) - fmaxf(ax1, bx1), 0.f);
    float ih = fmaxf(fminf(ay2, by2) - fmaxf(ay1, by1), 0.f);
    float inter = iw * ih;
    float uni = (ax2 - ax1) * (ay2 - ay1) + (bx2 - bx1) * (by2 - by1) - inter + 1e-7f;
    return inter / uni;
}

// ---- workspace init: cellTgt = -1, partials = 0 --------------------------
__global__ void k_init(int* __restrict__ cellTgt, int nCell,
                       float* __restrict__ part, int nPartF) {
    int i = blockIdx.x * blockDim.x + threadIdx.x;
    if (i < nCell)  cellTgt[i] = -1;
    if (i < nPartF) part[i] = 0.f;
}

// ---- target building: last-write-wins via atomicMax on order -------------
__global__ void k_build(const float* __restrict__ targets, int* __restrict__ cellTgt,
                        int scale, int H, int W) {
    int i = blockIdx.x * blockDim.x + threadIdx.x;
    if (i >= NB * NT) return;
    const float* t = targets + (size_t)i * CH;
    if (!(t[4] > 0.f)) return;                       // valid mask
    float cx = clamp01(t[0]), cy = clamp01(t[1]);
    float w  = clamp01(t[2]), h  = clamp01(t[3]);
    int best = 0; float bestIou = -1.f;
    for (int a = 0; a < NA; ++a) {                   // strict '>' => first max wins
        float aw = c_anchors[scale][a][0] / IMGF;
        float ah = c_anchors[scale][a][1] / IMGF;
        float v = iou_ccwh(0.5f, 0.5f, aw, ah, cx, cy, w, h);
        if (v > bestIou) { bestIou = v; best = a; }
    }
    int gxi = (int)(cx * (float)W); gxi = gxi < 0 ? 0 : (gxi > W - 1 ? W - 1 : gxi);
    int gyi = (int)(cy * (float)H); gyi = gyi < 0 ? 0 : (gyi > H - 1 ? H - 1 : gyi);
    int b = i / NT;
    int flat = ((b * NA + best) * H + gyi) * W + gxi;
    atomicMax(&cellTgt[flat], i);                    // order = b*NT + t = i
}

// ---- main loss: one thread per cell; negatives read only channel 4 -------
__global__ void __launch_bounds__(256)
k_loss(const float* __restrict__ pred, const float* __restrict__ targets,
       const int* __restrict__ cellTgt, float* __restrict__ part,
       int scale, int H, int W, int nCell) {
    int cell = blockIdx.x * 256 + threadIdx.x;
    float coord = 0.f, bpos = 0.f, bneg = 0.f, cls = 0.f, np = 0.f;
    if (cell < nCell) {
        const float* p = pred + (size_t)cell * CH;
        int tOrd = cellTgt[cell];
        float pc = sigf(p[4]);
        pc = fminf(fmaxf(pc, 1e-10f), 1.f - 1e-10f);
        if (tOrd >= 0) {
            np = 1.f;
            bpos = -logf(pc);
            int gx = cell % W;
            int gy = (cell / W) % H;
            int a  = (cell / (W * H)) % NA;
            float aw = c_anchors[scale][a][0], ah = c_anchors[scale][a][1];
            float bx = clamp01((sigf(p[0]) + (float)gx) / (float)W);
            float by = clamp01((sigf(p[1]) + (float)gy) / (float)H);
            float tw = fminf(fmaxf(p[2], -10.f), 10.f);
            float th = fminf(fmaxf(p[3], -10.f), 10.f);
            float bw = clamp01(expf(tw) * aw / IMGF);
            float bh = clamp01(expf(th) * ah / IMGF);
            const float* tg = targets + (size_t)tOrd * CH;
            float tx = clamp01(tg[0]), ty = clamp01(tg[1]);
            float tww = clamp01(tg[2]), thh = clamp01(tg[3]);
            coord = 1.f - iou_ccwh(bx, by, bw, bh, tx, ty, tww, thh);
            for (int c = 0; c < NC; ++c) {
                float q = sigf(p[5 + c]);
                q = fminf(fmaxf(q, 1e-10f), 1.f - 1e-10f);
                float tc = clamp01(tg[5 + c]);
                cls -= tc * logf(q) + (1.f - tc) * log1pf(-q);
            }
        } else {
            bneg = -log1pf(-pc);
        }
    }
    // deterministic block reduction: wave32 xor-butterfly, then LDS
    for (int m = 16; m > 0; m >>= 1) {
        coord += __shfl_xor(coord, m, 32);
        bpos  += __shfl_xor(bpos,  m, 32);
        bneg  += __shfl_xor(bneg,  m, 32);
        cls   += __shfl_xor(cls,   m, 32);
        np    += __shfl_xor(np,    m, 32);
    }
    __shared__ float red[8][5];
    int lane = threadIdx.x & 31, wv = threadIdx.x >> 5;
    if (lane == 0) {
        red[wv][0] = coord; red[wv][1] = bpos; red[wv][2] = bneg;
        red[wv][3] = cls;   red[wv][4] = np;
    }
    __syncthreads();
    if (threadIdx.x == 0) {
        float s0 = 0.f, s1 = 0.f, s2 = 0.f, s3 = 0.f, s4 = 0.f;
        for (int w2 = 0; w2 < 8; ++w2) {
            s0 += red[w2][0]; s1 += red[w2][1]; s2 += red[w2][2];
            s3 += red[w2][3]; s4 += red[w2][4];
        }
        float* row = part + (size_t)blockIdx.x * 16;
        row[0] = s0; row[1] = s1; row[2] = s2; row[3] = s3; row[4] = s4;
        for (int c = 5; c < 16; ++c) row[c] = 0.f;
    }
}

// ---- final: WMMA-based deterministic cross-block column-sum --------------
// A = ones(16x4), B = 4 partial rows (4x16). D[m][n] = sum_k B[k][n], so the
// column sums accumulate in C across the loop regardless of the K<->slot map.
__device__ __forceinline__ void wmma_colsum(const float* __restrict__ part,
                                            int pbRows, int lane, float* S) {
    v2f a; a[0] = 1.f; a[1] = 1.f;
    v8f acc = {};
    int half = lane >> 4, col = lane & 15;
    for (int base = 0; base < pbRows; base += 4) {
        v2f b;
        b[0] = part[(size_t)(base + half)     * 16 + col];
        b[1] = part[(size_t)(base + 2 + half) * 16 + col];
        acc = __builtin_amdgcn_wmma_f32_16x16x4_f32(false, a, false, b,
                                                    (short)0, acc, false, false);
    }
    float v = acc[0];               // row M=0 (all rows identical): col sum at lane n
    for (int c = 0; c < 5; ++c) S[c] = __shfl(v, c, 32);
}

__global__ void k_final(const float* __restrict__ part0, int pb0, int nc0,
                        const float* __restrict__ part1, int pb1, int nc1,
                        const float* __restrict__ part2, int pb2, int nc2,
                        float* __restrict__ out) {
    int lane = threadIdx.x;          // blockDim == 32, EXEC all ones for WMMA
    float S0[5], S1[5], S2[5];
    wmma_colsum(part0, pb0, lane, S0);
    wmma_colsum(part1, pb1, lane, S1);
    wmma_colsum(part2, pb2, lane, S2);
    if (lane == 0) {
        float coord = 0.f, conf = 0.f, cls = 0.f;
        float ncf[3] = { (float)nc0, (float)nc1, (float)nc2 };
        float* Ss[3] = { S0, S1, S2 };
        for (int s = 0; s < 3; ++s) {
            float* S = Ss[s];
            float np = S[4];
            float nneg = ncf[s] - np;
            float dp = fmaxf(np, 1.f);
            coord += (np > 0.f) ? S[0] / dp : 0.f;
            conf  += S[1] / dp + 0.5f * S[2] / fmaxf(nneg, 1.f);
            cls   += (np > 0.f) ? S[3] / fmaxf(np * (float)NC, 1.f) : 0.f;
        }
        float total = 5.f * coord + conf + 1.f * cls;
        out[0] = total; out[1] = coord; out[2] = conf; out[3] = cls;
    }
}

extern "C" void kernel_launch(void* const* d_in, const int* in_sizes, int n_in,
                              void* d_out, int out_size, void* d_ws, size_t ws_size,
                              hipStream_t stream) {
    (void)in_sizes; (void)n_in; (void)out_size; (void)ws_size;
    const float* pred0   = (const float*)d_in[0];
    const float* pred1   = (const float*)d_in[1];
    const float* pred2   = (const float*)d_in[2];
    const float* targets = (const float*)d_in[3];
    float* out = (float*)d_out;

    const int nc0 = NB * NA * 76 * 76;   // 277248
    const int nc1 = NB * NA * 38 * 38;   // 69312
    const int nc2 = NB * NA * 19 * 19;   // 17328
    const int nb0 = (nc0 + 255) / 256, nb1 = (nc1 + 255) / 256, nb2 = (nc2 + 255) / 256;
    const int pb0 = (nb0 + 3) & ~3, pb1 = (nb1 + 3) & ~3, pb2 = (nb2 + 3) & ~3;

    char* ws = (char*)d_ws;
    int* ct0 = (int*)ws;
    int* ct1 = ct0 + nc0;
    int* ct2 = ct1 + nc1;
    size_t ctBytes = ((size_t)(nc0 + nc1 + nc2) * sizeof(int) + 255) & ~(size_t)255;
    float* part0 = (float*)(ws + ctBytes);
    float* part1 = part0 + (size_t)pb0 * 16;
    float* part2 = part1 + (size_t)pb1 * 16;

    int nCellTot = nc0 + nc1 + nc2;
    int nPartF = (pb0 + pb1 + pb2) * 16;
    int initN = nCellTot > nPartF ? nCellTot : nPartF;
    k_init<<<(initN + 255) / 256, 256, 0, stream>>>(ct0, nCellTot, part0, nPartF);

    int tThreads = NB * NT;
    k_build<<<(tThreads + 255) / 256, 256, 0, stream>>>(targets, ct0, 0, 76, 76);
    k_build<<<(tThreads + 255) / 256, 256, 0, stream>>>(targets, ct1, 1, 38, 38);
    k_build<<<(tThreads + 255) / 256, 256, 0, stream>>>(targets, ct2, 2, 19, 19);

    k_loss<<<nb0, 256, 0, stream>>>(pred0, targets, ct0, part0, 0, 76, 76, nc0);
    k_loss<<<nb1, 256, 0, stream>>>(pred1, targets, ct1, part1, 1, 38, 38, nc1);
    k_loss<<<nb2, 256, 0, stream>>>(pred2, targets, ct2, part2, 2, 19, 19, nc2);

    k_final<<<1, 32, 0, stream>>>(part0, pb0, nc0, part1, pb1, nc1, part2, pb2, nc2, out);
}